// CDA_Subnet_73538430042842
// MI455X (gfx1250) — compile-verified
//
#include <hip/hip_runtime.h>

#define TAU    20.0f
#define V_TH   1.0f
#define LR     0.001f
#define NNEUR  1024
#define NUM_IN 2048

typedef __attribute__((ext_vector_type(2))) float v2f;
typedef __attribute__((ext_vector_type(8))) float v8f;

// ---------------------------------------------------------------------------
// Stage 1: LIF update. 1024 threads, thread j owns neuron j.
// Column reads W[i*N + j] are fully coalesced across the wave; the spike test
// is wave-uniform (same address for all lanes) so zero rows are skipped.
// ---------------------------------------------------------------------------
__global__ void lif_kernel(const float* __restrict__ in_spk,
                           const float* __restrict__ prev_spk,
                           const float* __restrict__ pot,
                           const float* __restrict__ W_in,
                           const float* __restrict__ W_rec,
                           float* __restrict__ out_spk,   // d_out[0..1023]
                           float* __restrict__ ws_spk) {  // scratch copy
    const int j = blockIdx.x * blockDim.x + threadIdx.x;  // 0..1023
    float acc = 0.0f;
    for (int i = 0; i < NUM_IN; ++i) {
        const float s = in_spk[i];                 // uniform -> scalar load
        if (s != 0.0f) acc += s * W_in[i * NNEUR + j];
    }
    for (int i = 0; i < NNEUR; ++i) {
        const float s = prev_spk[i];
        if (s != 0.0f) acc += s * W_rec[i * NNEUR + j];
    }
    const float p   = pot[j];
    const float v   = p + (-p / TAU + acc);        // DT = 1
    const float spk = (v >= V_TH) ? 1.0f : 0.0f;
    out_spk[j] = spk;
    ws_spk[j]  = spk;
}

// ---------------------------------------------------------------------------
// Stage 2: h = relu(combined @ pW1.T + pb1).  64 blocks x 128 threads,
// block o computes one hidden unit via LDS tree reduction. 512 KB total.
// ---------------------------------------------------------------------------
__global__ void hidden_kernel(const float* __restrict__ prev_spk,
                              const float* __restrict__ cur_spk,
                              const float* __restrict__ pW1,
                              const float* __restrict__ pb1,
                              float* __restrict__ h_out) {
    __shared__ float red[128];
    const int o = blockIdx.x;
    const int t = threadIdx.x;
    const float* row = pW1 + o * (2 * NNEUR);
    float acc = 0.0f;
    for (int i = t; i < 2 * NNEUR; i += 128) {
        const float c = (i < NNEUR) ? prev_spk[i] : cur_spk[i - NNEUR];
        acc += c * row[i];
    }
    red[t] = acc;
    __syncthreads();
    for (int s = 64; s > 0; s >>= 1) {
        if (t < s) red[t] += red[t + s];
        __syncthreads();
    }
    if (t == 0) {
        const float v = red[0] + pb1[o];
        h_out[o] = v > 0.0f ? v : 0.0f;
    }
}

// ---------------------------------------------------------------------------
// Stage 3: policy GEMV on the matrix pipe + structural plasticity.
// Each wave owns 16 output indices (one 16-element tile of N^2):
//   B tile: B[k][n] = pW2[(rowBase+n)*64 + k]   (16 pW2 rows, K=64)
//   A tile: h[k-chunk] replicated across all 16 M rows
// => every C row = the 16 dot products, one per lane of C VGPR0.
// Two accumulators: prune rows [idx] and genesis rows [N^2 + idx].
// pW2 is streamed once with non-temporal loads (512 MB > 192 MB L2).
//
// VGPR layouts per CDNA5 ISA 7.12.2 (A 16x4 f32: VGPR0 = K0|K2, VGPR1 = K1|K3
// across lane halves; B 4x16 mirrors it), so lane L with n = L&15,
// khalf = L>>4 loads the float2 at column offset kk*4 + khalf*2.
// ---------------------------------------------------------------------------
__global__ void policy_update_kernel(const float* __restrict__ pW2,
                                     const float* __restrict__ pb2,
                                     const float* __restrict__ h,
                                     const float* __restrict__ W_rec,
                                     float* __restrict__ out_W) {
    const int lane  = threadIdx.x & 31;
    const int wave  = blockIdx.x * (blockDim.x >> 5) + (threadIdx.x >> 5);
    const long rowBase = (long)wave * 16;          // 16 outputs per wave
    const int  n     = lane & 15;
    const int  khalf = lane >> 4;                  // 0 or 1
    const long NN    = (long)NNEUR * NNEUR;

    const float* pRow = pW2 + (rowBase + n) * 64 + khalf * 2;        // prune row
    const float* gRow = pW2 + (NN + rowBase + n) * 64 + khalf * 2;   // genesis row
    const float* hPtr = h + khalf * 2;

    v8f cp = {};
    v8f cg = {};
#pragma unroll
    for (int kk = 0; kk < 16; ++kk) {              // K = 64 in steps of 4
        const v2f a  = *(const v2f*)(hPtr + kk * 4);
        const v2f bp = __builtin_nontemporal_load((const v2f*)(pRow + kk * 4));
        const v2f bg = __builtin_nontemporal_load((const v2f*)(gRow + kk * 4));
        cp = __builtin_amdgcn_wmma_f32_16x16x4_f32(false, a, false, bp,
                                                   (short)0, cp, false, false);
        cg = __builtin_amdgcn_wmma_f32_16x16x4_f32(false, a, false, bg,
                                                   (short)0, cg, false, false);
    }

    if (lane < 16) {                               // lanes 16-31 hold duplicates
        const long idx = rowBase + n;
        const float pol_p = cp[0] + pb2[idx];
        const float pol_g = cg[0] + pb2[NN + idx];
        const float sp = 1.0f / (1.0f + __expf(-pol_p));
        const float sg = 1.0f / (1.0f + __expf(-pol_g));
        const int r = (int)(idx >> 10);
        const int c = (int)(idx & (NNEUR - 1));
        float w = W_rec[idx] - LR * sp + LR * sg;
        w = (w > 0.0f) ? w : 0.0f;
        out_W[idx] = (r == c) ? 0.0f : w;
    }
}

extern "C" void kernel_launch(void* const* d_in, const int* in_sizes, int n_in,
                              void* d_out, int out_size, void* d_ws, size_t ws_size,
                              hipStream_t stream) {
    const float* in_spk   = (const float*)d_in[0];  // [2048]
    const float* prev_spk = (const float*)d_in[1];  // [1024]
    const float* pot      = (const float*)d_in[2];  // [1024]
    const float* W_in     = (const float*)d_in[3];  // [2048*1024]
    const float* W_rec    = (const float*)d_in[4];  // [1024*1024]
    const float* pW1      = (const float*)d_in[5];  // [64*2048]
    const float* pb1      = (const float*)d_in[6];  // [64]
    const float* pW2      = (const float*)d_in[7];  // [2*1024*1024*64]
    const float* pb2      = (const float*)d_in[8];  // [2*1024*1024]

    float* out_spikes = (float*)d_out;              // [1024]
    float* out_Wrec   = (float*)d_out + NNEUR;      // [1024*1024]

    float* ws_spk = (float*)d_ws;                   // [1024]
    float* ws_h   = (float*)d_ws + NNEUR;           // [64]

    // Stage 1: LIF update (1024 neurons)
    lif_kernel<<<NNEUR / 256, 256, 0, stream>>>(in_spk, prev_spk, pot,
                                                W_in, W_rec, out_spikes, ws_spk);

    // Stage 2: hidden layer (64 units)
    hidden_kernel<<<64, 128, 0, stream>>>(prev_spk, ws_spk, pW1, pb1, ws_h);

    // Stage 3: policy GEMV + plasticity.  N^2/16 tiles, one per wave,
    // 8 waves per 256-thread block -> exact grid, EXEC all-ones for WMMA.
    const int waves  = (NNEUR * NNEUR) / 16;        // 65536
    const int blocks = waves / 8;                   // 8192
    policy_update_kernel<<<blocks, 256, 0, stream>>>(pW2, pb2, ws_h,
                                                     W_rec, out_Wrec);
}